// StateHistory_63058709840328
// MI455X (gfx1250) — compile-verified
//
#include <hip/hip_runtime.h>
#include <math.h>
#include <stdint.h>

#define NUM_E   200000
#define HID     128
#define N_EDGES 600000

typedef float v4f __attribute__((ext_vector_type(4)));

// CDNA5 hardware TANH transcendental if clang exposes it.
#if __has_builtin(__builtin_amdgcn_tanhf)
  #define FAST_TANH(x) __builtin_amdgcn_tanhf(x)
#elif __has_builtin(__builtin_amdgcn_tanh_f32)
  #define FAST_TANH(x) __builtin_amdgcn_tanh_f32(x)
#else
  #define FAST_TANH(x) tanhf(x)
#endif

// LDS byte offset of a generic pointer to a __shared__ object.
__device__ __forceinline__ unsigned lds_offset(const void* p) {
  return (unsigned)(uintptr_t)(__attribute__((address_space(3))) const void*)p;
}

// CDNA5 async gather: 16B per lane, global -> LDS, tracked by ASYNCcnt.
__device__ __forceinline__ void async_gather_b128(unsigned lds_off, const void* gaddr) {
  asm volatile("global_load_async_to_lds_b128 %0, %1, off"
               :: "v"(lds_off), "v"(gaddr) : "memory");
}
__device__ __forceinline__ void wait_async_le(int n) {
  // immediate-only SOPP; n is a compile-time constant at all call sites
  asm volatile("s_wait_asynccnt %0" :: "i"(n) : "memory");
}
__device__ __forceinline__ void wait_ds0() {
  asm volatile("s_wait_dscnt 0x0" ::: "memory");
}

// ---------------------------------------------------------------------------
// Kernel 1: zero the aggregation buffer (lives in d_out's `tmp` half).
// ---------------------------------------------------------------------------
__global__ void zero_agg_kernel(v4f* __restrict__ agg, int n4) {
  int i = blockIdx.x * blockDim.x + threadIdx.x;
  int stride = gridDim.x * blockDim.x;
  v4f z = {0.f, 0.f, 0.f, 0.f};
  for (; i < n4; i += stride) agg[i] = z;
}

// ---------------------------------------------------------------------------
// Kernel 2: per-edge gather + scale + scatter-add with an async-to-LDS
// software pipeline. One wave per edge per stage; each stage is a single
// global_load_async_to_lds_b128 (32 lanes x 16B = one 512B embedding row).
// DEPTH rows are kept in flight per wave (ASYNCcnt), hiding the random-gather
// latency without holding in-flight data in VGPRs. Atomic f32 adds land in
// the L2-resident (102MB < 192MB) accumulator.
// ---------------------------------------------------------------------------
#define WAVES_PER_BLOCK 8
#define DEPTH 8   // outstanding async row-gathers per wave

__global__ void edge_scatter_kernel(const float* __restrict__ emb,
                                    const long long* __restrict__ eid,  // [2, E] int64
                                    const float* __restrict__ ew,
                                    float* __restrict__ agg) {
  __shared__ float buf[WAVES_PER_BLOCK][DEPTH][HID];  // 32 KB

  const int lane   = threadIdx.x & 31;
  const int warp   = threadIdx.x >> 5;
  const int wave   = blockIdx.x * WAVES_PER_BLOCK + warp;
  const int nwaves = gridDim.x * WAVES_PER_BLOCK;

  // Issue the async gather for pipeline stage k (clamped so tail refills are
  // valid memory accesses; their results are simply never consumed).
  auto issue = [&](int k) {
    int e = wave + k * nwaves;
    if (e >= N_EDGES) e = 0;
    long long src = eid[e];
    const float* g = emb + src * (long long)HID + lane * 4;
    async_gather_b128(lds_offset(&buf[warp][k & (DEPTH - 1)][lane * 4]), g);
  };

  // Prologue: fill the pipeline.
  #pragma unroll
  for (int s = 0; s < DEPTH; ++s) issue(s);

  // Steady state: wait-oldest, consume, refill.
  for (int k = 0;; ++k) {
    int e = wave + k * nwaves;
    if (e >= N_EDGES) break;

    long long dst = eid[N_EDGES + e];
    float w = ew[e];

    wait_async_le(DEPTH - 1);   // oldest in-flight gather has landed in LDS
    v4f v = *reinterpret_cast<const v4f*>(&buf[warp][k & (DEPTH - 1)][lane * 4]);

    float* a = agg + dst * (long long)HID + (long long)(lane * 4);
    unsafeAtomicAdd(a + 0, v.x * w);   // native global_atomic_add_f32
    unsafeAtomicAdd(a + 1, v.y * w);
    unsafeAtomicAdd(a + 2, v.z * w);
    unsafeAtomicAdd(a + 3, v.w * w);

    wait_ds0();        // LDS read of this slot complete -> safe to refill (WAR)
    issue(k + DEPTH);  // reuse the slot we just consumed
  }
}

// ---------------------------------------------------------------------------
// Kernel 3: fused tail. Reads agg (from the tmp region) + emb, computes
//   h   = tanh(agg*w_diag + emb*loop_w + bias)
//   tmp = res * h          (overwrites agg in place — same v4f per thread)
//   out = emb + tmp        (streamed non-temporal store)
// ---------------------------------------------------------------------------
__global__ void fused_tail_kernel(const float* __restrict__ emb,
                                  const float* __restrict__ w_diag,
                                  const float* __restrict__ loop_w,
                                  const float* __restrict__ bias,
                                  const float* __restrict__ res,
                                  float* __restrict__ out,
                                  float* __restrict__ tmp /* holds agg on entry */) {
  const float r = res[0];
  const int n4 = NUM_E * HID / 4;
  const int stride = gridDim.x * blockDim.x;

  for (int i = blockIdx.x * blockDim.x + threadIdx.x; i < n4; i += stride) {
    const int col4 = i & (HID / 4 - 1);

    v4f w = reinterpret_cast<const v4f*>(w_diag)[col4];
    v4f l = reinterpret_cast<const v4f*>(loop_w)[col4];
    v4f b = reinterpret_cast<const v4f*>(bias)[col4];
    v4f a = reinterpret_cast<v4f*>(tmp)[i];
    v4f x = reinterpret_cast<const v4f*>(emb)[i];

    v4f h;
    h.x = FAST_TANH(fmaf(a.x, w.x, fmaf(x.x, l.x, b.x)));
    h.y = FAST_TANH(fmaf(a.y, w.y, fmaf(x.y, l.y, b.y)));
    h.z = FAST_TANH(fmaf(a.z, w.z, fmaf(x.z, l.z, b.z)));
    h.w = FAST_TANH(fmaf(a.w, w.w, fmaf(x.w, l.w, b.w)));

    v4f t = r * h;
    v4f o = x + t;

    reinterpret_cast<v4f*>(tmp)[i] = t;                              // tmp
    __builtin_nontemporal_store(o, reinterpret_cast<v4f*>(out) + i); // out
  }
}

// ---------------------------------------------------------------------------
extern "C" void kernel_launch(void* const* d_in, const int* in_sizes, int n_in,
                              void* d_out, int out_size, void* d_ws, size_t ws_size,
                              hipStream_t stream) {
  (void)in_sizes; (void)n_in; (void)out_size; (void)d_ws; (void)ws_size;

  const float*     emb    = (const float*)d_in[0];
  const long long* eid    = (const long long*)d_in[1];  // int64 [2, E]
  const float*     ew     = (const float*)d_in[2];
  const float*     w_diag = (const float*)d_in[3];
  const float*     loop_w = (const float*)d_in[4];
  const float*     bias   = (const float*)d_in[5];
  const float*     res    = (const float*)d_in[6];

  float* out = (float*)d_out;                 // first  N*HID floats
  float* tmp = out + (size_t)NUM_E * HID;     // second N*HID floats (agg, then tmp)

  const int n4 = NUM_E * HID / 4;

  // 1) zero the accumulator (tmp region)
  {
    int threads = 256;
    int blocks  = (n4 + threads - 1) / threads;
    zero_agg_kernel<<<blocks, threads, 0, stream>>>(
        reinterpret_cast<v4f*>(tmp), n4);
  }

  // 2) edge scatter: 4096 blocks * 8 waves = 32,768 waves, ~18 edges/wave,
  //    each keeping DEPTH=8 async row-gathers in flight.
  {
    int threads = 32 * WAVES_PER_BLOCK;
    int blocks  = 4096;
    edge_scatter_kernel<<<blocks, threads, 0, stream>>>(emb, eid, ew, tmp);
  }

  // 3) fused tail
  {
    int threads = 256;
    int blocks  = (n4 + threads - 1) / threads;
    fused_tail_kernel<<<blocks, threads, 0, stream>>>(
        emb, w_diag, loop_w, bias, res, out, tmp);
  }
}